// HEPTKernel_85555748536856
// MI455X (gfx1250) — compile-verified
//
#include <hip/hip_runtime.h>

typedef float v2f __attribute__((ext_vector_type(2)));
typedef float v8f __attribute__((ext_vector_type(8)));
typedef int   v4i __attribute__((vector_size(4 * sizeof(int))));

#define H   8
#define BB  4
#define NN  2048
#define DD  64
#define EPSV 0.0009765625f   // 2^-10

#define QTILE 16            // query rows per wave
#define WAVES 4             // waves per block (128 threads)
#define KTILE 16            // keys per iteration
#define LSTR  68            // padded dword stride for K/V LDS tiles (bank = 4*row+col)
#define WSTR  20            // padded dword stride for W bounce tile
#define TILEF (KTILE * LSTR)   // dwords per K or V tile in LDS

#if __has_builtin(__builtin_amdgcn_global_load_async_to_lds_b128)
#define HAVE_ASYNC_LDS 1
#else
#define HAVE_ASYNC_LDS 0
#endif

__device__ __forceinline__ void async_copy16(const float* gsrc, float* lds) {
#if HAVE_ASYNC_LDS
    __builtin_amdgcn_global_load_async_to_lds_b128(
        (__attribute__((address_space(1))) v4i*)gsrc,
        (__attribute__((address_space(3))) v4i*)lds,
        0, 0);
#else
    float4 tmp = *(const float4*)gsrc;   // short live range: no cross-loop spill
    *(float4*)lds = tmp;
#endif
}

__device__ __forceinline__ void wait_async_all() {
#if HAVE_ASYNC_LDS
#if __has_builtin(__builtin_amdgcn_s_wait_asynccnt)
    __builtin_amdgcn_s_wait_asynccnt(0);
#else
    asm volatile("s_wait_asynccnt 0" ::: "memory");
#endif
#endif
}

__global__ __launch_bounds__(128) void hept_attn_kernel(
    const float* __restrict__ Qg, const float* __restrict__ Kg,
    const float* __restrict__ Vg, const int* __restrict__ pm,
    float* __restrict__ Og)
{
    // double-buffered K+V tiles (K tile at offset 0, V tile at offset TILEF)
    __shared__ __align__(16) float ldsKV[2][2 * TILEF];
    __shared__ __align__(16) float ldsW[WAVES * 16 * WSTR];

    const int tid  = threadIdx.x;
    const int wave = tid >> 5;
    const int lane = tid & 31;
    const int ln   = lane & 15;   // A-row / B-col / D-col index
    const int hi   = lane >> 4;   // half-wave select

    const int hb    = blockIdx.x;          // head*B + b
    const int head  = hb >> 2;
    const int b     = hb & 3;
    const int qbase = (blockIdx.y * WAVES + wave) * QTILE;

    const size_t baseHB = ((size_t)head * (BB * NN) + (size_t)b * NN) * DD;
    const float* Qh  = Qg + baseHB;
    const float* Kh  = Kg + baseHB;
    const float* Vh  = Vg + baseHB;
    const int*   pmb = pm + b * NN;

    // ---- cooperative staging assignment: 128 threads x 4 x b128 = 8KB/tile-pair ----
    const int sel  = tid >> 6;         // 0 -> K, 1 -> V
    const int srow = (tid & 63) >> 2;  // row within 16-row tile
    const int sseg = tid & 3;          // 16-dword segment
    const float* gsel   = (sel ? Vh : Kh) + (size_t)srow * DD + sseg * 16;
    const int    selofs = sel * TILEF + srow * LSTR + sseg * 16;

    // ---- Q tile resident in registers, WMMA-A layout; ||q||^2 on the fly ----
    // A element (row=ln, k=4*kk + 2*hi + j) in qa[kk][j]
    v2f qa[16];
    float q2own = 0.0f;
    {
        const float* qrow = Qh + (size_t)(qbase + ln) * DD + 2 * hi;
        #pragma unroll
        for (int kk = 0; kk < 16; ++kk) {
            v2f v = *(const v2f*)(qrow + 4 * kk);
            qa[kk] = v;
            q2own += v.x * v.x + v.y * v.y;      // this lane holds half the row
        }
        q2own += __shfl_xor(q2own, 16, 32);      // lane^16 holds the other half
    }
    float q2r[8];                                 // per accumulator row r (row = r + 8*hi)
    #pragma unroll
    for (int r = 0; r < 8; ++r)
        q2r[r] = __shfl(q2own, r + 8 * hi, 32);

    // ---- prologue: async-stage tile 0 into buffer 0 ----
    #pragma unroll
    for (int i = 0; i < 4; ++i)
        async_copy16(gsel + 4 * i, &ldsKV[0][selofs + 4 * i]);
    wait_async_all();
    __syncthreads();

    v8f oacc0 = {}, oacc1 = {}, oacc2 = {}, oacc3 = {};
    v8f wsum  = {};
    float* myW = ldsW + wave * (16 * WSTR);

    const int NT = NN / KTILE;   // 128 key tiles
    for (int t = 0; t < NT; ++t) {
        // async-stage tile t+1 into the other buffer while computing tile t
        if (t + 1 < NT) {
            const float* gsrc = gsel + (size_t)(t + 1) * KTILE * DD;
            float* dst = &ldsKV[(t + 1) & 1][selofs];
            #pragma unroll
            for (int i = 0; i < 4; ++i)
                async_copy16(gsrc + 4 * i, dst + 4 * i);
        }
        if (t + 2 < NT)
            __builtin_prefetch(gsel + (size_t)(t + 2) * KTILE * DD, 0, 1);

        const float* bK = &ldsKV[t & 1][0];
        const float* bV = &ldsKV[t & 1][TILEF];
        const int kb = t * KTILE;

        // ---- S = Q * K^T : 16 f32 WMMAs (2 accumulators for ILP) ----
        v8f s0 = {}, s1 = {};
        float k2part = 0.0f;
        const float* krow = bK + ln * LSTR + 2 * hi;   // B element (col=ln, k=4kk+2hi+j)
        #pragma unroll
        for (int kk = 0; kk < 16; kk += 2) {
            v2f b0 = *(const v2f*)(krow + 4 * kk);
            v2f b1 = *(const v2f*)(krow + 4 * (kk + 1));
            k2part = fmaf(b0.x, b0.x, k2part);
            k2part = fmaf(b0.y, b0.y, k2part);
            k2part = fmaf(b1.x, b1.x, k2part);
            k2part = fmaf(b1.y, b1.y, k2part);
            s0 = __builtin_amdgcn_wmma_f32_16x16x4_f32(false, qa[kk],     false, b0,
                                                       (short)0, s0, false, false);
            s1 = __builtin_amdgcn_wmma_f32_16x16x4_f32(false, qa[kk + 1], false, b1,
                                                       (short)0, s1, false, false);
        }
        v8f s = s0 + s1;

        const float k2c = k2part + __shfl_xor(k2part, 16, 32);  // ||k||^2 for key kb+ln
        const float mk  = (pmb[kb + ln] != 0) ? 1.0f : 0.0f;
        const float cb  = -0.5f * k2c;                          // per-column bias

        // ---- W = mask * exp(S - 0.5 q2 - 0.5 k2); arg <= 0 so no overflow ----
        // D layout: s[r] at (row = r+8*hi, col = ln). Bounce to LDS for A-layout reads.
        #pragma unroll
        for (int r = 0; r < 8; ++r) {
            float w = mk * __expf(s[r] + cb - 0.5f * q2r[r]);
            wsum[r] += w;                                       // elementwise; reduce at end
            myW[(r + 8 * hi) * WSTR + ln] = w;
        }
        // same-wave LDS RAW: hardware dscnt wait, no barrier needed

        // ---- O += W * V : A = W (via LDS transpose), B = V tile ----
        #pragma unroll
        for (int c = 0; c < 4; ++c) {
            v2f wa = *(const v2f*)(myW + ln * WSTR + 4 * c + 2 * hi);
            const float* v0 = bV + (4 * c + 2 * hi + 0) * LSTR + ln;
            const float* v1 = bV + (4 * c + 2 * hi + 1) * LSTR + ln;
            #pragma unroll
            for (int g = 0; g < 4; ++g) {
                v2f vb;
                vb.x = v0[g * 16];
                vb.y = v1[g * 16];
                v8f* oa = (g == 0) ? &oacc0 : (g == 1) ? &oacc1 : (g == 2) ? &oacc2 : &oacc3;
                *oa = __builtin_amdgcn_wmma_f32_16x16x4_f32(false, wa, false, vb,
                                                            (short)0, *oa, false, false);
            }
        }

        wait_async_all();        // this wave's tile-(t+1) transfers have landed
        __syncthreads();         // all waves' transfers visible; all done reading tile t
    }

    // ---- row sums: reduce wsum over the 16 lanes of each half-wave ----
    float rs[8];
    #pragma unroll
    for (int r = 0; r < 8; ++r) {
        float v = wsum[r];
        v += __shfl_xor(v, 1, 32);
        v += __shfl_xor(v, 2, 32);
        v += __shfl_xor(v, 4, 32);
        v += __shfl_xor(v, 8, 32);
        rs[r] = v;
    }

    // ---- out(row) = mask_q * O(row) / (rowsum + EPS) ----
    #pragma unroll
    for (int r = 0; r < 8; ++r) {
        const int qrow = qbase + r + 8 * hi;
        const float mq  = (pmb[qrow] != 0) ? 1.0f : 0.0f;
        const float inv = mq / (rs[r] + EPSV);
        float* orow = Og + baseHB + (size_t)qrow * DD;
        orow[0 * 16 + ln] = oacc0[r] * inv;
        orow[1 * 16 + ln] = oacc1[r] * inv;
        orow[2 * 16 + ln] = oacc2[r] * inv;
        orow[3 * 16 + ln] = oacc3[r] * inv;
    }
}

extern "C" void kernel_launch(void* const* d_in, const int* in_sizes, int n_in,
                              void* d_out, int out_size, void* d_ws, size_t ws_size,
                              hipStream_t stream) {
    (void)in_sizes; (void)n_in; (void)out_size; (void)d_ws; (void)ws_size;
    const float* Q  = (const float*)d_in[0];
    const float* K  = (const float*)d_in[1];
    const float* V  = (const float*)d_in[2];
    const int*   pm = (const int*)d_in[3];
    float* out = (float*)d_out;

    dim3 grid(H * BB, NN / (QTILE * WAVES));   // (32, 32)
    hept_attn_kernel<<<grid, WAVES * 32, 0, stream>>>(Q, K, V, pm, out);
}